// GaussianEBM_82205674045713
// MI455X (gfx1250) — compile-verified
//
#include <hip/hip_runtime.h>
#include <cstdint>

#define NPIX 784
#define NSIDE 28
#define ROWS_PER_WG 8
#define BLOCK 256
// LDS layout (floats): [0, 6272) = 8 rows of x, [6272, 6304) = zero tail,
// [6304, 8656) = coefficients cd(784) | cr(784) | cdn(784)
#define LDS_X_TAIL (ROWS_PER_WG * NPIX)          // 6272
#define LDS_COEF   (ROWS_PER_WG * NPIX + 32)     // 6304
#define LDS_TOTAL  (LDS_COEF + 3 * NPIX)         // 8656 floats = 34.6 KB

typedef __attribute__((ext_vector_type(4))) unsigned int tdm_g0_t;
typedef __attribute__((ext_vector_type(8))) int          tdm_g1_t;
typedef __attribute__((ext_vector_type(4))) int          tdm_g2_t;

// ---------------------------------------------------------------------------
// Kernel 1: fold precision*mask into 3 stencil band coefficients (2352 floats)
// cd[i] = A[i,i];  cr[i] = A[i,i+1]+A[i+1,i];  cdn[i] = A[i,i+28]+A[i+28,i]
// ---------------------------------------------------------------------------
__global__ void coeff_kernel(const float* __restrict__ P,
                             const float* __restrict__ M,
                             float* __restrict__ ws) {
    int i = blockIdx.x * blockDim.x + threadIdx.x;
    if (i >= NPIX) return;
    int gx = i % NSIDE, gy = i / NSIDE;
    long r = (long)i * NPIX;
    float cd = P[r + i] * M[r + i];
    float cr = 0.0f, cdn = 0.0f;
    if (gx < NSIDE - 1)
        cr = P[r + i + 1] * M[r + i + 1] + P[r + NPIX + i] * M[r + NPIX + i];
    if (gy < NSIDE - 1)
        cdn = P[r + i + NSIDE] * M[r + i + NSIDE]
            + P[r + (long)NSIDE * NPIX + i] * M[r + (long)NSIDE * NPIX + i];
    ws[i]            = cd;
    ws[NPIX + i]     = cr;
    ws[2 * NPIX + i] = cdn;
}

// ---------------------------------------------------------------------------
// Kernel 2: one wave32 per batch row; 8 rows per WG staged via TDM into LDS.
// TDM issued first so the 25 KB DMA overlaps coefficient staging.
// ---------------------------------------------------------------------------
__global__ __launch_bounds__(BLOCK)
void ebm_kernel(const float* __restrict__ x,
                const float* __restrict__ ws,
                const float* __restrict__ logZp,
                float* __restrict__ out) {
    __shared__ __align__(16) float smem[LDS_TOTAL];
    const int tid  = threadIdx.x;
    const int lane = tid & 31;
    const int wv   = tid >> 5;

    float* sc = smem + LDS_COEF;

    if (wv == 0) {
        // --- Tensor Data Mover: DMA an 8 x 784 f32 tile (25 KB) into LDS ---
        unsigned long long ga =
            (unsigned long long)(uintptr_t)(x + (size_t)blockIdx.x * ROWS_PER_WG * NPIX);
        unsigned int lds_off = (unsigned int)(uintptr_t)(void*)smem; // low 32b of flat = LDS byte offset

        tdm_g0_t g0;
        g0.x = 1u;                                             // count=1 valid descriptor
        g0.y = lds_off;                                        // lds_addr (bytes)
        g0.z = (unsigned int)(ga & 0xFFFFFFFFu);               // global_addr[31:0]
        g0.w = (unsigned int)((ga >> 32) & 0x01FFFFFFu)        // global_addr[56:32]
             | (2u << 30);                                     // type = 2 ("image")

        tdm_g1_t g1;
        g1[0] = 0x00020000;                                    // wg_mask=0, data_size=2 (4B)
        g1[1] = (int)((NPIX & 0xFFFF) << 16);                  // tensor_dim0[15:0]=784
        g1[2] = (int)(ROWS_PER_WG << 16);                      // tensor_dim1[15:0]=8
        g1[3] = (int)((NPIX & 0xFFFF) << 16);                  // tile_dim0=784
        g1[4] = ROWS_PER_WG;                                   // tile_dim1=8, tile_dim2=0
        g1[5] = NPIX;                                          // tensor_dim0_stride=784
        g1[6] = (int)(((NPIX * ROWS_PER_WG) & 0xFFFF) << 16);  // tensor_dim1_stride lo
        g1[7] = 0;

        tdm_g2_t gz4  = {0, 0, 0, 0};                          // groups 2/3 unused (2D tensor)
        tdm_g1_t gz8  = {0, 0, 0, 0, 0, 0, 0, 0};              // extra octet (clang-23 form)

        // clang-23 / therock-10.0 form: 6 args (g0, g1, g2, g3, extra, cpol)
        __builtin_amdgcn_tensor_load_to_lds(g0, g1, gz4, gz4, gz8, 0);
    }

    // Stage the 2352 stencil coefficients into LDS while the DMA is in flight
    // (b128 loads from L2-resident ws, b128 LDS stores).
    for (int k = tid * 4; k < 3 * NPIX; k += BLOCK * 4)
        *(float4*)(sc + k) = *(const float4*)(ws + k);
    // Zero the 32-float tail so k+1 / k+28 reads past row 7 are finite
    // (their coefficients are exactly 0, so contribution is 0).
    if (tid < 32) smem[LDS_X_TAIL + tid] = 0.0f;

    if (wv == 0) {
        __builtin_amdgcn_s_wait_tensorcnt(0);
    }
    __syncthreads();

    // --- Stencil quadratic form: this wave owns one batch row ---
    const float* sx = smem + wv * NPIX;
    const float* cd = sc;
    const float* cr = sc + NPIX;
    const float* cn = sc + 2 * NPIX;

    float acc = 0.0f;
    #pragma unroll
    for (int it = 0; it < 7; ++it) {
        const int k = it * 128 + lane * 4;       // 196 float4 per row
        if (k < NPIX) {
            const float4 v  = *(const float4*)(sx + k);
            const float4 dn = *(const float4*)(sx + k + NSIDE);
            const float  nx = sx[k + 4];
            const float4 c0 = *(const float4*)(cd + k);
            const float4 c1 = *(const float4*)(cr + k);
            const float4 c2 = *(const float4*)(cn + k);
            acc = fmaf(v.x, fmaf(c0.x, v.x, fmaf(c1.x, v.y, c2.x * dn.x)), acc);
            acc = fmaf(v.y, fmaf(c0.y, v.y, fmaf(c1.y, v.z, c2.y * dn.y)), acc);
            acc = fmaf(v.z, fmaf(c0.z, v.z, fmaf(c1.z, v.w, c2.z * dn.z)), acc);
            acc = fmaf(v.w, fmaf(c0.w, v.w, fmaf(c1.w, nx,  c2.w * dn.w)), acc);
        }
    }

    // wave32 reduction
    #pragma unroll
    for (int off = 16; off > 0; off >>= 1) acc += __shfl_xor(acc, off, 32);

    if (lane == 0) {
        const float lz = logZp[0];
        out[(size_t)blockIdx.x * ROWS_PER_WG + wv] = fmaf(-0.5f, acc, -lz);
    }
}

// ---------------------------------------------------------------------------
extern "C" void kernel_launch(void* const* d_in, const int* in_sizes, int n_in,
                              void* d_out, int out_size, void* d_ws, size_t ws_size,
                              hipStream_t stream) {
    const float* x  = (const float*)d_in[0];   // [B, 784] f32
    const float* P  = (const float*)d_in[1];   // [784, 784] f32
    const float* lz = (const float*)d_in[2];   // scalar f32
    const float* M  = (const float*)d_in[3];   // [784, 784] f32
    float* out = (float*)d_out;                // [B] f32
    float* ws  = (float*)d_ws;                 // >= 2352 floats used

    const int B = in_sizes[0] / NPIX;          // 65536

    coeff_kernel<<<(NPIX + 255) / 256, 256, 0, stream>>>(P, M, ws);
    ebm_kernel<<<B / ROWS_PER_WG, BLOCK, 0, stream>>>(x, ws, lz, out);
}